// q6_3_48473000903102
// MI455X (gfx1250) — compile-verified
//
#include <hip/hip_runtime.h>

#define VOCAB 100000
#define EMB   300
#define BATCH 2048
#define SEQ   256

typedef __attribute__((ext_vector_type(2))) float v2f;
typedef __attribute__((ext_vector_type(8))) float v8f;

// Phase 1: s[v] = dot(W_emb[v, :], fc_w[:])  via V_WMMA_F32_16X16X4_F32.
//
// Orientation: A (16x4) holds fc_w[k..k+3] in row M=0 only (lane 0 -> K0,K1;
// lane 16 -> K2,K3; all other lanes zero via cndmask on unconditionally-loaded
// LDS values -> no EXEC juggling in the loop). B (4x16) holds a 16-row tile of
// W_emb: lane n (n=lane&15) carries row m0+n, K-half selected by lane>>4 --
// an 8-byte global load per lane, K-sequential within each row.
//
// Result: D row M=0 = the 16 dot products, living in c[0] across lanes 0..15
// -> single coalesced 64B store.
//
// Grid is exact (625 blocks * 10 waves * 16 rows = 100000) -> EXEC all-1s at WMMA.
__global__ void __launch_bounds__(320) emb_dot_wmma(const float* __restrict__ W,
                                                    const float* __restrict__ fcw,
                                                    float* __restrict__ s) {
    __shared__ float sf[EMB];
    for (int i = threadIdx.x; i < EMB; i += 320) sf[i] = fcw[i];
    __syncthreads();

    const int  lane   = threadIdx.x & 31;
    const int  waveId = threadIdx.x >> 5;
    const int  m0     = (blockIdx.x * 10 + waveId) * 16;  // 16 rows per wave-tile
    const int  col    = lane & 15;                        // B: N index (row within tile)
    const int  khalf  = (lane >> 4) * 2;                  // K pair {0,1} vs {2,3}
    const bool row0   = (col == 0);                       // A: M==0 lives in lanes 0,16
    const float* wrow = W + (size_t)(m0 + col) * EMB + khalf;  // 8-byte aligned

    v8f c = {};
#pragma unroll 5
    for (int k = 0; k < EMB; k += 4) {                    // 75 K-steps
        // B-operand: streaming W_emb tile (global_load_b64, K-sequential per row)
        v2f b = *(const v2f*)(wrow + k);
        // A-operand: fc_w chunk, loaded unconditionally (ds_load_b64), masked
        // to row M=0 with plain selects (v_cndmask, no exec save/restore).
        float t0 = sf[k + khalf];
        float t1 = sf[k + khalf + 1];
        v2f a;
        a.x = row0 ? t0 : 0.0f;
        a.y = row0 ? t1 : 0.0f;
        // 8 args: (neg_a, A, neg_b, B, c_mod, C, reuse_a, reuse_b)
        c = __builtin_amdgcn_wmma_f32_16x16x4_f32(false, a, false, b,
                                                  (short)0, c, false, false);
    }

    // D layout: (M=0, N=n) sits in VGPR c[0], lanes 0..15 -> coalesced store.
    if (lane < 16)
        s[m0 + lane] = c[0];
}

// Phase 2: one wave per batch item. Coalesced token loads, gathers from the
// L2-resident 400KB s-table, wave32 xor-shuffle reduction, sigmoid on lane 0.
__global__ void __launch_bounds__(256) pool_sigmoid(const int* __restrict__ tokens,
                                                    const int* __restrict__ lengths,
                                                    const float* __restrict__ s,
                                                    const float* __restrict__ fcb,
                                                    float* __restrict__ out) {
    const int lane = threadIdx.x & 31;
    const int item = (int)((blockIdx.x * 256u + threadIdx.x) >> 5);
    if (item >= BATCH) return;

    const int  len  = lengths[item];
    const int* trow = tokens + item * SEQ;

    float acc = 0.0f;
    for (int l = lane; l < len; l += 32)
        acc += s[trow[l]];

#pragma unroll
    for (int off = 16; off > 0; off >>= 1)
        acc += __shfl_xor(acc, off, 32);

    if (lane == 0) {
        float z = acc / (float)len + fcb[0];
        out[item] = 1.0f / (1.0f + __expf(-z));
    }
}

extern "C" void kernel_launch(void* const* d_in, const int* in_sizes, int n_in,
                              void* d_out, int out_size, void* d_ws, size_t ws_size,
                              hipStream_t stream) {
    const int*   tokens  = (const int*)  d_in[0];
    const int*   lengths = (const int*)  d_in[1];
    const float* W       = (const float*)d_in[2];
    const float* fcw     = (const float*)d_in[3];
    const float* fcb     = (const float*)d_in[4];
    float*       out     = (float*)d_out;
    float*       s       = (float*)d_ws;   // 100000 floats = 400 KB scratch

    // 625 blocks * 10 waves/block * 16 rows/wave = 100000 rows exactly.
    emb_dot_wmma<<<625, 320, 0, stream>>>(W, fcw, s);
    // 2048 items, 8 waves (items) per 256-thread block.
    pool_sigmoid<<<256, 256, 0, stream>>>(tokens, lengths, s, fcb, out);
}